// DHCF_1_66185446031942
// MI455X (gfx1250) — compile-verified
//
#include <hip/hip_runtime.h>

#define NN 100000
#define DD 128
#define KEEP 0.9f
#define LDS_STRIDE 132   // 128 + 4 pad: lanes 0-15 reading a column hit distinct banks

typedef __attribute__((ext_vector_type(2))) float v2f;
typedef __attribute__((ext_vector_type(8))) float v8f;

__device__ __forceinline__ float hash01(unsigned x) {
    x ^= x >> 16; x *= 0x7feb352dU;
    x ^= x >> 15; x *= 0x846ca68bU;
    x ^= x >> 16;
    return (float)(x >> 8) * (1.0f / 16777216.0f);
}

// ---------------- kernel 1: zero the agg accumulator ----------------
__global__ __launch_bounds__(256) void dhcf_zero(float* __restrict__ p, int n4) {
    int i = blockIdx.x * blockDim.x + threadIdx.x;
    if (i < n4) {
        float4 z = make_float4(0.f, 0.f, 0.f, 0.f);
        ((float4*)p)[i] = z;
    }
}

// ---------------- kernel 2: SpMM, agg[row] += vals * e[col] ----------------
// one wave32 per nonzero; lane l handles cols [4l, 4l+3] (float4 coalesced)
__global__ __launch_bounds__(256) void dhcf_spmm(const int* __restrict__ row,
                                                 const int* __restrict__ col,
                                                 const float* __restrict__ vals,
                                                 const float* __restrict__ embed,
                                                 float* __restrict__ agg, int E) {
    int wave = (blockIdx.x * blockDim.x + threadIdx.x) >> 5;
    int lane = threadIdx.x & 31;
    if (wave >= E) return;
    int r = row[wave];
    int c = col[wave];
    float v = vals[wave];
    float4 ev = *(const float4*)(embed + (size_t)c * DD + lane * 4);
    float* dst = agg + (size_t)r * DD + lane * 4;
    unsafeAtomicAdd(dst + 0, v * ev.x);
    unsafeAtomicAdd(dst + 1, v * ev.y);
    unsafeAtomicAdd(dst + 2, v * ev.z);
    unsafeAtomicAdd(dst + 3, v * ev.w);
}

// ---------------- kernel 3: residual + dropout + GEMM(WMMA f32) + bias + leaky ----------------
// 4 waves / block; each wave owns one 16-row tile. N = 6250 * 16 exactly.
__global__ __launch_bounds__(128) void dhcf_fused(const float* __restrict__ agg,
                                                  const float* __restrict__ embed,
                                                  const float* __restrict__ fcw,
                                                  const float* __restrict__ fcb,
                                                  float* __restrict__ out) {
    __shared__ float lds[4][16 * LDS_STRIDE];
    const int wv   = threadIdx.x >> 5;
    const int lane = threadIdx.x & 31;
    const int tile = blockIdx.x * 4 + wv;
    const bool active = tile < (NN / 16);
    const int r0 = tile * 16;

    if (active) {
        // build m1 = dropout(agg + e) into LDS; stream e to out[:, :128]
        for (int i = 0; i < 16; ++i) {
            size_t off = (size_t)(r0 + i) * DD + lane * 4;
            float4 a4 = *(const float4*)(agg + off);
            float4 e4 = *(const float4*)(embed + off);
            *(float4*)(out + (size_t)(r0 + i) * (2 * DD) + lane * 4) = e4;  // concat part 1
            float m[4] = {a4.x + e4.x, a4.y + e4.y, a4.z + e4.z, a4.w + e4.w};
            unsigned base = (unsigned)off;
#pragma unroll
            for (int j = 0; j < 4; ++j)
                m[j] = (hash01(base + j) < KEEP) ? m[j] * (1.0f / KEEP) : 0.0f;
            float* dst = &lds[wv][i * LDS_STRIDE + lane * 4];
            dst[0] = m[0]; dst[1] = m[1]; dst[2] = m[2]; dst[3] = m[3];
        }
    }
    __syncthreads();
    if (!active) return;

    const int half = lane >> 4;       // lane-half: K offset 2h, C/D row offset 8h
    const int lpos = lane & 15;       // A row / B col / C col
    const float* arow = &lds[wv][lpos * LDS_STRIDE];

    v8f acc[8];
#pragma unroll
    for (int c = 0; c < 8; ++c) {
        v8f z = {0.f, 0.f, 0.f, 0.f, 0.f, 0.f, 0.f, 0.f};
        acc[c] = z;
    }

    // x1[16,128] = m1[16,128] @ fc_w.T : B[k][n] = fc_w[n][k]
#pragma unroll 4
    for (int kb = 0; kb < DD; kb += 4) {
        const int k = kb + 2 * half;
        v2f a = *(const v2f*)(arow + k);                          // ds_load_b64
#pragma unroll
        for (int c = 0; c < 8; ++c) {
            v2f b = *(const v2f*)(fcw + (size_t)(c * 16 + lpos) * DD + k);
            acc[c] = __builtin_amdgcn_wmma_f32_16x16x4_f32(
                false, a, false, b, (short)0, acc[c], false, false);
        }
    }

    // epilogue: bias + leaky_relu(0.2), write out[:, 128:256]
#pragma unroll
    for (int c = 0; c < 8; ++c) {
        float bias = fcb[c * 16 + lpos];
#pragma unroll
        for (int v = 0; v < 8; ++v) {
            int rowi = v + 8 * half;
            float x = acc[c][v] + bias;
            x = (x >= 0.f) ? x : 0.2f * x;
            out[(size_t)(r0 + rowi) * (2 * DD) + DD + c * 16 + lpos] = x;
        }
    }
}

extern "C" void kernel_launch(void* const* d_in, const int* in_sizes, int n_in,
                              void* d_out, int out_size, void* d_ws, size_t ws_size,
                              hipStream_t stream) {
    // inputs: x, row, col, vals, embed_w, fc_w, fc_b
    const int*   row   = (const int*)d_in[1];
    const int*   col   = (const int*)d_in[2];
    const float* vals  = (const float*)d_in[3];
    const float* embed = (const float*)d_in[4];
    const float* fcw   = (const float*)d_in[5];
    const float* fcb   = (const float*)d_in[6];
    float* out = (float*)d_out;
    float* agg = (float*)d_ws;                    // N*D floats = 51.2 MB scratch
    const int E = in_sizes[1];

    const int n4 = (NN * DD) / 4;
    dhcf_zero<<<(n4 + 255) / 256, 256, 0, stream>>>(agg, n4);

    dhcf_spmm<<<(E + 7) / 8, 256, 0, stream>>>(row, col, vals, embed, agg, E);

    const int tiles = NN / 16;                    // 6250
    dhcf_fused<<<(tiles + 3) / 4, 128, 0, stream>>>(agg, embed, fcw, fcb, out);
}